// ConvRelu_90881507983641
// MI455X (gfx1250) — compile-verified
//
#include <hip/hip_runtime.h>
#include <hip/hip_bf16.h>

typedef float v2f __attribute__((ext_vector_type(2)));
typedef float v8f __attribute__((ext_vector_type(8)));

// ---------------------------------------------------------------------------
// Kernel 1: zero workspace (agg + out_deg + in_deg), grid-stride.
// ---------------------------------------------------------------------------
__global__ void gc_zero_kernel(float* __restrict__ p, long long n) {
    long long i = (long long)blockIdx.x * blockDim.x + threadIdx.x;
    long long stride = (long long)gridDim.x * blockDim.x;
    for (; i < n; i += stride) p[i] = 0.0f;
}

// ---------------------------------------------------------------------------
// Kernel 2: degree histograms via float atomics (global_atomic_add_f32).
// ---------------------------------------------------------------------------
__global__ void gc_degree_kernel(const int* __restrict__ src,
                                 const int* __restrict__ dst,
                                 float* __restrict__ out_deg,
                                 float* __restrict__ in_deg,
                                 int n_edges) {
    int e = blockIdx.x * blockDim.x + threadIdx.x;
    if (e >= n_edges) return;
    atomicAdd(&out_deg[src[e]], 1.0f);
    atomicAdd(&in_deg[dst[e]], 1.0f);
}

// ---------------------------------------------------------------------------
// Kernel 3: edge scatter.  One wave32 per edge; lane handles 4 features
// (float4 load -> fully coalesced 512B per edge).  agg[dst] += feat[src]*ns.
// Both 51.2MB buffers are L2-resident (192MB L2); atomic RMW at L2.
// ---------------------------------------------------------------------------
__global__ __launch_bounds__(256) void gc_scatter_kernel(
        const float* __restrict__ feature,
        const int* __restrict__ src,
        const int* __restrict__ dst,
        const float* __restrict__ out_deg,
        float* __restrict__ agg,
        int n_edges) {
    int e = blockIdx.x * 8 + (threadIdx.x >> 5);
    if (e >= n_edges) return;
    int lane = threadIdx.x & 31;
    int s = src[e];
    int d = dst[e];
    float ns = __frsqrt_rn(fmaxf(out_deg[s], 1.0f));
    const float4* f4 = (const float4*)(feature + (size_t)s * 128);
    float4 v = f4[lane];
    float* ag = agg + (size_t)d * 128 + lane * 4;
    atomicAdd(ag + 0, v.x * ns);
    atomicAdd(ag + 1, v.y * ns);
    atomicAdd(ag + 2, v.z * ns);
    atomicAdd(ag + 3, v.w * ns);
}

// ---------------------------------------------------------------------------
// Kernel 4: fused (agg * norm_dst) @ W + b, ReLU   via V_WMMA_F32_16X16X4_F32.
// Block = 256 threads = 8 waves; block owns 16 output rows; wave w owns
// output columns [16w, 16w+16).  K=128 -> 32 WMMA issues per wave.
//
// The 16x128 A tile (scaled by norm_dst) is staged once in LDS and shared by
// all 8 waves (8x cut in A read traffic).  LDS row stride = 132 floats so the
// ds_load_b64 in the hot loop (lane m, half h -> banks {4m+2h, 4m+2h+1}) hits
// all 64 banks exactly once -> conflict-free.
//
// f32 WMMA layouts (ISA 7.12.2):
//   A 16x4 : lane l -> M = l&15 ; VGPR0/1 = K = 2*(l>>4), 2*(l>>4)+1
//   B 4x16 : lane l -> N = l&15 ; VGPR0/1 = K = 2*(l>>4), 2*(l>>4)+1
//   C 16x16: VGPR r -> M = r + 8*(l>>4), N = l&15
// ---------------------------------------------------------------------------
#define LDS_STRIDE 132

__global__ __launch_bounds__(256) void gc_gemm_relu_kernel(
        const float* __restrict__ agg,
        const float* __restrict__ in_deg,
        const float* __restrict__ W,
        const float* __restrict__ bias,
        float* __restrict__ out,
        int n_nodes) {
    __shared__ __align__(16) float As[16 * LDS_STRIDE];   // 8.25 KB

    const int t    = threadIdx.x;
    const int row0 = blockIdx.x * 16;

    // ---- stage A tile (x norm_dst) into LDS: thread t -> row t>>4, 8 floats
    {
        const int fr = t >> 4;              // 0..15
        const int fc = (t & 15) * 8;        // 0,8,...,120
        int grow  = row0 + fr;
        int growc = grow < n_nodes ? grow : n_nodes - 1;
        float nd  = __frsqrt_rn(fmaxf(in_deg[growc], 1.0f));
        const float4* s4 = (const float4*)(agg + (size_t)growc * 128 + fc);
        float4 u0 = s4[0];
        float4 u1 = s4[1];
        float* dl = &As[fr * LDS_STRIDE + fc];
        *(float4*)(dl + 0) = make_float4(u0.x * nd, u0.y * nd, u0.z * nd, u0.w * nd);
        *(float4*)(dl + 4) = make_float4(u1.x * nd, u1.y * nd, u1.z * nd, u1.w * nd);
    }
    __syncthreads();

    const int lane = t & 31;
    const int wave = t >> 5;             // column tile 0..7
    const int n0   = wave * 16;
    const int mn   = lane & 15;          // M for A-read, N for B-load/store
    const int half = lane >> 4;          // 0 or 1
    const int koff = 2 * half;

    const float bv = bias[n0 + mn];
    const float* lrow = &As[mn * LDS_STRIDE + koff];

    v8f acc = {};
#pragma unroll 4
    for (int k = 0; k < 128; k += 4) {
        v2f a  = *(const v2f*)(lrow + k);        // ds_load_b64, conflict-free
        v2f bm;
        bm.x = W[(size_t)(k + koff + 0) * 128 + n0 + mn];
        bm.y = W[(size_t)(k + koff + 1) * 128 + n0 + mn];
        // v_wmma_f32_16x16x4_f32: (neg_a, A, neg_b, B, c_mod, C, reuse_a, reuse_b)
        acc = __builtin_amdgcn_wmma_f32_16x16x4_f32(
            false, a, false, bm, (short)0, acc, false, false);
    }

    // ---- epilogue: bias + ReLU + store (guard is block-uniform)
    float* obase = out + (size_t)row0 * 128 + n0 + mn;
    if (row0 + 16 <= n_nodes) {
#pragma unroll
        for (int r = 0; r < 8; ++r) {
            int orow = r + 8 * half;
            obase[(size_t)orow * 128] = fmaxf(acc[r] + bv, 0.0f);
        }
    } else {
#pragma unroll
        for (int r = 0; r < 8; ++r) {
            int orow = row0 + r + 8 * half;
            if (orow < n_nodes)
                out[(size_t)orow * 128 + n0 + mn] = fmaxf(acc[r] + bv, 0.0f);
        }
    }
}

// ---------------------------------------------------------------------------
// Launch: inputs = feature[N*128], src[E], dst[E], W[128*128], b[128]
// d_out = relu((D_in^-1/2 * scatter(D_out^-1/2 * feat)) @ W + b)  [N,128] f32
// ---------------------------------------------------------------------------
extern "C" void kernel_launch(void* const* d_in, const int* in_sizes, int n_in,
                              void* d_out, int out_size, void* d_ws, size_t ws_size,
                              hipStream_t stream) {
    const float* feature = (const float*)d_in[0];
    const int*   src     = (const int*)d_in[1];
    const int*   dst     = (const int*)d_in[2];
    const float* W       = (const float*)d_in[3];
    const float* bias    = (const float*)d_in[4];

    const int n_nodes = in_sizes[0] / 128;   // 100000
    const int n_edges = in_sizes[1];         // 600000

    float* agg     = (float*)d_ws;                       // n_nodes*128 floats
    float* out_deg = agg + (size_t)n_nodes * 128;        // n_nodes floats
    float* in_deg  = out_deg + n_nodes;                  // n_nodes floats
    const long long zero_n = (long long)n_nodes * 128 + 2LL * n_nodes;

    // 1) zero accumulator + degree buffers
    gc_zero_kernel<<<2048, 256, 0, stream>>>((float*)d_ws, zero_n);

    // 2) degrees
    gc_degree_kernel<<<(n_edges + 255) / 256, 256, 0, stream>>>(
        src, dst, out_deg, in_deg, n_edges);

    // 3) scatter-add (1 wave per edge, 8 edges per 256-thread block)
    gc_scatter_kernel<<<(n_edges + 7) / 8, 256, 0, stream>>>(
        feature, src, dst, out_deg, agg, n_edges);

    // 4) fused normalize + GEMM (WMMA f32) + bias + ReLU
    gc_gemm_relu_kernel<<<(n_nodes + 15) / 16, 256, 0, stream>>>(
        agg, in_deg, W, bias, (float*)d_out, n_nodes);
}